// ProbabilisticFastRCNNOutputLayers_84602265797277
// MI455X (gfx1250) — compile-verified
//
#include <hip/hip_runtime.h>
#include <hip/hip_bf16.h>
#include <math.h>

// ---------------------------------------------------------------------------
// ProbabilisticFastRCNNOutputLayers for MI455X (gfx1250).
//   out[N,90] = [scores(81) | pred_boxes(4) | cov(4) | mean_iou(1)]
// GEMM (N=65536, C=1024, 89 outputs) on V_WMMA_F32_16X16X4_F32 (exact fp32),
// with double-buffered GLOBAL_LOAD_ASYNC_TO_LDS_B128 staging (ASYNCcnt).
// ---------------------------------------------------------------------------

typedef __attribute__((ext_vector_type(2))) float v2f;
typedef __attribute__((ext_vector_type(8))) float v8f;

#define N_ROWS   65536
#define C_DIM    1024
#define N_CT     6          // 6 column tiles of 16 -> 96 cols (89 used)
#define KSTEPS   256        // 1024 / 4
#define LDS_STR  260        // 256 + 4 pad: 16B-aligned rows, conflict-free frags
#define OUT_STR  90

// ---------------------------------------------------------------------------
// Pack W_cls/W_box/W_cov into per-wave B fragments for wmma_f32_16x16x4_f32.
//   VGPR r, lane L : B[K = r + 2*(L>>4)][N = L&15]  with B[k][j] = W[j][k]
// Linear index:  Wfrag[((ct*KSTEPS + ks)*32 + lane)] = {W[j][k], W[j][k+1]}
//   j = ct*16 + (lane&15),  k = ks*4 + 2*(lane>>4)
// ---------------------------------------------------------------------------
__global__ __launch_bounds__(256) void pfrcnn_pack(
    const float* __restrict__ Wcls, const float* __restrict__ Wbox,
    const float* __restrict__ Wcov, const float* __restrict__ bcls,
    const float* __restrict__ bbox, const float* __restrict__ bcov,
    v2f* __restrict__ Wfrag, float* __restrict__ bias)
{
    const int t = blockIdx.x * blockDim.x + threadIdx.x;
    if (t < N_CT * KSTEPS * 32) {
        const int lane = t & 31;
        const int ks   = (t >> 5) & (KSTEPS - 1);
        const int ct   = t >> 13;                 // / (KSTEPS*32)
        const int j    = ct * 16 + (lane & 15);
        const int k    = ks * 4 + ((lane >> 4) << 1);
        v2f v; v.x = 0.f; v.y = 0.f;
        const float* src = nullptr;
        if (j < 81)       src = Wcls + j * C_DIM + k;
        else if (j < 85)  src = Wbox + (j - 81) * C_DIM + k;
        else if (j < 89)  src = Wcov + (j - 85) * C_DIM + k;
        if (src) { v.x = src[0]; v.y = src[1]; }
        Wfrag[t] = v;
    }
    if (t < 96) {
        float b = 0.f;
        if (t < 81)      b = bcls[t];
        else if (t < 85) b = bbox[t - 81];
        else if (t < 89) b = bcov[t - 85];
        bias[t] = b;
    }
}

// ---------------------------------------------------------------------------
// Async global->LDS stage of one 16x256-float chunk (1024 x b128 issues,
// spread over 192 threads). No VGPR round trip; tracked by ASYNCcnt.
// LDS operand = low 32 bits of the flat shared pointer.
// ---------------------------------------------------------------------------
__device__ __forceinline__ void stage_chunk_async(
    const float* __restrict__ feat, int r0, int kc, float* sbuf, int tid)
{
    for (int idx = tid; idx < 1024; idx += 192) {
        const int row = idx >> 6;               // 64 float4 per row
        const int c4  = idx & 63;
        const float* g =
            feat + (size_t)(r0 + row) * C_DIM + kc * 256 + c4 * 4;
        const unsigned ldsoff =
            (unsigned)(uintptr_t)(&sbuf[row * LDS_STR + c4 * 4]);
        asm volatile("global_load_async_to_lds_b128 %0, %1, off"
                     :: "v"(ldsoff),
                        "v"((unsigned long long)(uintptr_t)g)
                     : "memory");
    }
}

__device__ __forceinline__ void wait_async0() {
    asm volatile("s_wait_asynccnt 0" ::: "memory");
}

// ---------------------------------------------------------------------------
// GEMM: block = 16-row strip x 96 cols, 6 waves, wave ct owns cols 16ct..16ct+15.
// Double-buffered async LDS staging; 2 accumulators break the WMMA RAW chain.
// ---------------------------------------------------------------------------
__global__ __launch_bounds__(192) void pfrcnn_gemm(
    const float* __restrict__ feat, const v2f* __restrict__ Wfrag,
    const float* __restrict__ bias, float* __restrict__ out,
    float* __restrict__ lin)
{
    __shared__ float sA[2][16 * LDS_STR];

    const int tid  = threadIdx.x;
    const int lane = tid & 31;
    const int ct   = tid >> 5;                  // wave id == column tile
    const int r0   = blockIdx.x * 16;
    const int m    = lane & 15;                 // A row within tile
    const int q    = (lane >> 4) << 1;          // 0 or 2 (K sub-offset)

    v8f acc0 = {0.f, 0.f, 0.f, 0.f, 0.f, 0.f, 0.f, 0.f};
    v8f acc1 = {0.f, 0.f, 0.f, 0.f, 0.f, 0.f, 0.f, 0.f};

    const v2f* wp = Wfrag + (ct * KSTEPS) * 32 + lane;

    stage_chunk_async(feat, r0, 0, sA[0], tid);

    for (int kc = 0; kc < 4; ++kc) {
        wait_async0();          // own async writes for buf[kc&1] have landed
        __syncthreads();        // everyone's landed; buf[(kc+1)&1] free to reuse
        if (kc < 3)
            stage_chunk_async(feat, r0, kc + 1, sA[(kc + 1) & 1], tid);

        const float* sbuf = sA[kc & 1];
        const v2f* wk = wp + (kc * 64) * 32;
        #pragma unroll 4
        for (int kk = 0; kk < 64; kk += 2) {
            const v2f a0 = *(const v2f*)&sbuf[m * LDS_STR + kk * 4 + q];
            const v2f a1 = *(const v2f*)&sbuf[m * LDS_STR + (kk + 1) * 4 + q];
            const v2f b0 = wk[kk * 32];
            const v2f b1 = wk[(kk + 1) * 32];
            acc0 = __builtin_amdgcn_wmma_f32_16x16x4_f32(
                false, a0, false, b0, (short)0, acc0, false, false);
            acc1 = __builtin_amdgcn_wmma_f32_16x16x4_f32(
                false, a1, false, b1, (short)0, acc1, false, false);
        }
    }

    const v8f acc = acc0 + acc1;

    // Epilogue: C/D layout -> VGPR r holds M = r (lanes 0-15) / r+8 (lanes 16-31),
    // N = lane&15. Scores go to d_out (stride 90); cols 81..88 to workspace.
    const int n  = lane & 15;
    const int j  = ct * 16 + n;
    const int lh = (lane >> 4) << 3;            // 0 or 8
    const float bj = bias[j];
    #pragma unroll
    for (int r = 0; r < 8; ++r) {
        const int row = r0 + r + lh;
        const float v = acc[r] + bj;
        if (j < 81)
            out[(size_t)row * OUT_STR + j] = v;
        else if (j < 89)
            lin[(size_t)row * 8 + (j - 81)] = v;
    }
}

// ---------------------------------------------------------------------------
// Per-row box decode + 100-sample Monte-Carlo IoU. One wave32 per row;
// lanes parallelize samples (coalesced float4 eps reads), shfl_xor reduce.
// ---------------------------------------------------------------------------
__global__ __launch_bounds__(256) void pfrcnn_box(
    const float* __restrict__ prop, const float* __restrict__ eps,
    const float* __restrict__ lin, float* __restrict__ out)
{
    const int wave = (blockIdx.x * blockDim.x + threadIdx.x) >> 5;
    const int lane = threadIdx.x & 31;
    if (wave >= N_ROWS) return;
    const int row = wave;

    const float SC = 4.1351665567423560f;       // log(1000/16)

    // broadcast loads (same address across the wave)
    const float p0 = prop[row * 4 + 0], p1 = prop[row * 4 + 1];
    const float p2 = prop[row * 4 + 2], p3 = prop[row * 4 + 3];
    const float d0 = lin[row * 8 + 0],  d1 = lin[row * 8 + 1];
    const float d2 = lin[row * 8 + 2],  d3 = lin[row * 8 + 3];
    const float c0 = lin[row * 8 + 4],  c1 = lin[row * 8 + 5];
    const float c2 = lin[row * 8 + 6],  c3 = lin[row * 8 + 7];

    // proposal box
    const float x1 = p0 * 800.f, y1 = p1 * 800.f;
    const float w  = p2 * 300.f + 1.f, h = p3 * 300.f + 1.f;
    const float cx = x1 + 0.5f * w, cy = y1 + 0.5f * h;

    // apply_deltas + clip -> pred box
    const float dx = d0 * 0.1f, dy = d1 * 0.1f;
    const float dw = fminf(d2 * 0.2f, SC), dh = fminf(d3 * 0.2f, SC);
    const float pcx = dx * w + cx, pcy = dy * h + cy;
    const float pw  = expf(dw) * w, ph = expf(dh) * h;
    float b0 = pcx - 0.5f * pw, b1 = pcy - 0.5f * ph;
    float b2 = pcx + 0.5f * pw, b3 = pcy + 0.5f * ph;
    b0 = fminf(fmaxf(b0, 0.f), 1024.f);
    b1 = fminf(fmaxf(b1, 0.f), 1024.f);
    b2 = fminf(fmaxf(b2, 0.f), 1024.f);
    b3 = fminf(fmaxf(b3, 0.f), 1024.f);

    // cov = exp(clip(linear, -7, 7)); sigma = sqrt(cov)
    const float q0 = expf(fminf(fmaxf(c0, -7.f), 7.f));
    const float q1 = expf(fminf(fmaxf(c1, -7.f), 7.f));
    const float q2 = expf(fminf(fmaxf(c2, -7.f), 7.f));
    const float q3 = expf(fminf(fmaxf(c3, -7.f), 7.f));
    const float s0 = sqrtf(q0), s1 = sqrtf(q1), s2 = sqrtf(q2), s3 = sqrtf(q3);

    // base geometry of pred box
    const float bw = b2 - b0, bh = b3 - b1;
    const float bcx = b0 + 0.5f * bw, bcy = b1 + 0.5f * bh;
    const float barea = bw * bh;

    float sum = 0.f;
    for (int s = lane; s < 100; s += 32) {
        const float4 e = *(const float4*)(eps + ((size_t)row * 100 + s) * 4);
        const float sdx = e.x * s0 * 0.1f;
        const float sdy = e.y * s1 * 0.1f;
        const float sdw = fminf(e.z * s2 * 0.2f, SC);
        const float sdh = fminf(e.w * s3 * 0.2f, SC);
        const float scx = sdx * bw + bcx, scy = sdy * bh + bcy;
        const float sw  = expf(sdw) * bw, sh = expf(sdh) * bh;
        const float a0 = scx - 0.5f * sw, a1 = scy - 0.5f * sh;
        const float a2 = scx + 0.5f * sw, a3 = scy + 0.5f * sh;
        const float ltx = fmaxf(a0, b0), lty = fmaxf(a1, b1);
        const float rbx = fminf(a2, b2), rby = fminf(a3, b3);
        const float iw = fmaxf(rbx - ltx, 0.f), ih = fmaxf(rby - lty, 0.f);
        const float inter = iw * ih;
        const float area1 = (a2 - a0) * (a3 - a1);
        const float uni = area1 + barea - inter;
        sum += (inter > 0.f) ? inter / fmaxf(uni, 1e-12f) : 0.f;
    }
    // wave32 reduction
    for (int off = 16; off > 0; off >>= 1)
        sum += __shfl_xor(sum, off, 32);

    if (lane == 0) {
        float* o = out + (size_t)row * OUT_STR;
        o[81] = b0; o[82] = b1; o[83] = b2; o[84] = b3;
        o[85] = q0; o[86] = q1; o[87] = q2; o[88] = q3;
        o[89] = sum * 0.01f;
    }
}

// ---------------------------------------------------------------------------
extern "C" void kernel_launch(void* const* d_in, const int* in_sizes, int n_in,
                              void* d_out, int out_size, void* d_ws, size_t ws_size,
                              hipStream_t stream) {
    const float* feat = (const float*)d_in[0];   // [65536,1024]
    const float* prop = (const float*)d_in[1];   // [65536,4]
    const float* eps  = (const float*)d_in[2];   // [65536,100,4]
    const float* Wcls = (const float*)d_in[3];   // [81,1024]
    const float* bcls = (const float*)d_in[4];   // [81]
    const float* Wbox = (const float*)d_in[5];   // [4,1024]
    const float* bbox = (const float*)d_in[6];   // [4]
    const float* Wcov = (const float*)d_in[7];   // [4,1024]
    const float* bcov = (const float*)d_in[8];   // [4]
    float* out = (float*)d_out;                  // [65536,90]

    // workspace layout (floats): Wfrag[6*256*32*2] | bias[128] | lin[65536*8]
    float* ws    = (float*)d_ws;
    v2f*   Wfrag = (v2f*)ws;                       // 98304 floats
    float* bias  = ws + N_CT * KSTEPS * 32 * 2;    // 96 used, 128 reserved
    float* lin   = bias + 128;                     // 524288 floats

    pfrcnn_pack<<<(N_CT * KSTEPS * 32 + 255) / 256, 256, 0, stream>>>(
        Wcls, Wbox, Wcov, bcls, bbox, bcov, Wfrag, bias);

    pfrcnn_gemm<<<N_ROWS / 16, 192, 0, stream>>>(feat, Wfrag, bias, out, lin);

    pfrcnn_box<<<(N_ROWS * 32) / 256, 256, 0, stream>>>(prop, eps, lin, out);
}